// KGConv_72567767433688
// MI455X (gfx1250) — compile-verified
//
#include <hip/hip_runtime.h>
#include <math.h>

#define DEMB 64

typedef float v2f __attribute__((ext_vector_type(2)));
typedef float v8f __attribute__((ext_vector_type(8)));

#if defined(__AMDGCN__) && __has_builtin(__builtin_amdgcn_wmma_f32_16x16x4_f32)
#define USE_WMMA_F32 1
#else
#define USE_WMMA_F32 0
#endif

// ---------------------------------------------------------------------------
// Batched per-edge dot products via V_WMMA_F32_16X16X4_F32.
// One wave handles 16 edges. A = 16 gathered "left" rows (16x64),
// B^T = 16 gathered "right" rows; chain 16 K=4 WMMAs -> C = A*B (16x16),
// diag(C)[i] = dot(left_i, right_i). Full fp32.
//
// f32 A 16x4 layout: lanes 0-15 hold M=lane {K0,K1}; lanes 16-31 {K2,K3}.
// f32 B 4x16 layout: lanes 0-15 hold N=lane {K0,K1}; lanes 16-31 {K2,K3}.
// => lane (m = lane&15, h = lane>>4) loads elements {4k+2h, 4k+2h+1} of
//    edge (base+m)'s left row (A) and right row (B).
// C 16x16 f32: lanes 0-15: M=v, N=lane; lanes 16-31: M=v+8, N=lane-16.
// diag i<8 at (vgpr=i, lane=i); diag i>=8 at (vgpr=i-8, lane=i+16).
// ---------------------------------------------------------------------------

// KG edges: w1[e] = dot(entity[tail[e]], relation[etype[e]-1]) / 8
__global__ void kg_edge_logits(const float* __restrict__ ent,
                               const float* __restrict__ rel,
                               const int* __restrict__ tail,
                               const int* __restrict__ etype,
                               float* __restrict__ w1, int E) {
  int tid  = blockIdx.x * blockDim.x + threadIdx.x;
  int wave = tid >> 5;
  int lane = tid & 31;
  int base = wave * 16;
  if (base >= E) return;                 // wave-uniform; EXEC stays all-1s
  int m = lane & 15, h = lane >> 4;
  int e = base + m; if (e >= E) e = E - 1;   // clamp, keep EXEC full for WMMA
  const float* rowA = ent + (size_t)tail[e] * DEMB;
  const float* rowB = rel + (size_t)(etype[e] - 1) * DEMB;
#if USE_WMMA_F32
  v8f c = {0.f, 0.f, 0.f, 0.f, 0.f, 0.f, 0.f, 0.f};
#pragma unroll
  for (int k = 0; k < 16; ++k) {
    v2f a = *(const v2f*)(rowA + 4 * k + 2 * h);
    v2f b = *(const v2f*)(rowB + 4 * k + 2 * h);
    c = __builtin_amdgcn_wmma_f32_16x16x4_f32(false, a, false, b,
                                              (short)0, c, false, false);
  }
  float dot = 0.f; int di = 32;
#pragma unroll
  for (int v = 0; v < 8; ++v) {
    if (lane == v)      { dot = c[v]; di = v; }
    if (lane == v + 24) { dot = c[v]; di = v + 8; }
  }
  if (di < 16) {
    int eo = base + di;
    if (eo < E) w1[eo] = dot * 0.125f;
  }
#else
  float p = 0.f;
#pragma unroll
  for (int k = 0; k < 16; ++k) {
    v2f a = *(const v2f*)(rowA + 4 * k + 2 * h);
    v2f b = *(const v2f*)(rowB + 4 * k + 2 * h);
    p += a.x * b.x + a.y * b.y;
  }
  p += __shfl_xor(p, 16, 32);            // combine the two K-halves
  if (h == 0 && base + m < E) w1[base + m] = p * 0.125f;
#endif
}

// User interactions: att[i] = dot(user[u]*interact[t], entity[item])
__global__ void user_edge_logits(const float* __restrict__ ent,
                                 const float* __restrict__ usr,
                                 const float* __restrict__ inter,
                                 const int* __restrict__ uidx,
                                 const int* __restrict__ iidx,
                                 const int* __restrict__ itype,
                                 float* __restrict__ att, int n) {
  int tid  = blockIdx.x * blockDim.x + threadIdx.x;
  int wave = tid >> 5;
  int lane = tid & 31;
  int base = wave * 16;
  if (base >= n) return;
  int m = lane & 15, h = lane >> 4;
  int e = base + m; if (e >= n) e = n - 1;
  const float* rowU = usr   + (size_t)uidx[e]  * DEMB;
  const float* rowI = inter + (size_t)itype[e] * DEMB;
  const float* rowE = ent   + (size_t)iidx[e]  * DEMB;
#if USE_WMMA_F32
  v8f c = {0.f, 0.f, 0.f, 0.f, 0.f, 0.f, 0.f, 0.f};
#pragma unroll
  for (int k = 0; k < 16; ++k) {
    v2f u = *(const v2f*)(rowU + 4 * k + 2 * h);
    v2f t = *(const v2f*)(rowI + 4 * k + 2 * h);
    v2f a = u * t;                       // elementwise user*interact
    v2f b = *(const v2f*)(rowE + 4 * k + 2 * h);
    c = __builtin_amdgcn_wmma_f32_16x16x4_f32(false, a, false, b,
                                              (short)0, c, false, false);
  }
  float dot = 0.f; int di = 32;
#pragma unroll
  for (int v = 0; v < 8; ++v) {
    if (lane == v)      { dot = c[v]; di = v; }
    if (lane == v + 24) { dot = c[v]; di = v + 8; }
  }
  if (di < 16) {
    int eo = base + di;
    if (eo < n) att[eo] = dot;
  }
#else
  float p = 0.f;
#pragma unroll
  for (int k = 0; k < 16; ++k) {
    v2f u = *(const v2f*)(rowU + 4 * k + 2 * h);
    v2f t = *(const v2f*)(rowI + 4 * k + 2 * h);
    v2f b = *(const v2f*)(rowE + 4 * k + 2 * h);
    p += u.x * t.x * b.x + u.y * t.y * b.y;
  }
  p += __shfl_xor(p, 16, 32);
  if (h == 0 && base + m < n) att[base + m] = p;
#endif
}

// ---------------------------------------------------------------------------
// Segment softmax pieces + scatter + normalize
// ---------------------------------------------------------------------------
__global__ void fill_k(float* __restrict__ p, float v, int n) {
  int i = blockIdx.x * blockDim.x + threadIdx.x;
  if (i < n) p[i] = v;
}

// float atomic max via int/uint punning (valid incl. -inf init, mixed signs)
__global__ void seg_max_k(const float* __restrict__ w, const int* __restrict__ seg,
                          float* __restrict__ m, int n) {
  int i = blockIdx.x * blockDim.x + threadIdx.x;
  if (i < n) {
    float v = w[i];
    float* a = m + seg[i];
    if (v >= 0.f) atomicMax((int*)a, __float_as_int(v));
    else          atomicMin((unsigned int*)a, __float_as_uint(v));
  }
}

// w[i] <- exp(w[i]-m[seg]); s[seg] += w[i]
__global__ void seg_expsum_k(float* __restrict__ w, const int* __restrict__ seg,
                             const float* __restrict__ m, float* __restrict__ s, int n) {
  int i = blockIdx.x * blockDim.x + threadIdx.x;
  if (i < n) {
    int g = seg[i];
    float e = expf(w[i] - m[g]);
    w[i] = e;
    atomicAdd(s + g, e);
  }
}

// agg[seg[e]][d] += table[rowIdx[e]][d] * (ew[e]/s[seg[e]]), 64 threads/edge
__global__ void scatter_weighted_rows(const float* __restrict__ table,
                                      const int* __restrict__ rowIdx,
                                      const int* __restrict__ segIdx,
                                      const float* __restrict__ ew,
                                      const float* __restrict__ s,
                                      float* __restrict__ agg, int n) {
  long long t = (long long)blockIdx.x * blockDim.x + threadIdx.x;
  int e = (int)(t >> 6);
  int d = (int)(t & 63);
  if (e < n) {
    int g = segIdx[e];
    float wv = ew[e] / s[g];
    float val = table[(size_t)rowIdx[e] * DEMB + d];
    atomicAdd(agg + (size_t)g * DEMB + d, val * wv);
  }
}

// buf[row] <- buf[row]/max(||buf[row]||2, eps); res[row] += buf[row]
// one wave32 per row, float2 per lane, xor-shuffle reduction
__global__ void l2norm_residual(float* __restrict__ buf, float* __restrict__ res, int rows) {
  int tid  = blockIdx.x * blockDim.x + threadIdx.x;
  int wave = tid >> 5;
  int lane = tid & 31;
  if (wave >= rows) return;
  size_t off = (size_t)wave * DEMB + (size_t)lane * 2;
  v2f v = *(v2f*)(buf + off);
  float ss = v.x * v.x + v.y * v.y;
#pragma unroll
  for (int d = 16; d > 0; d >>= 1) ss += __shfl_xor(ss, d, 32);
  float inv = 1.f / fmaxf(sqrtf(ss), 1e-12f);
  v2f o; o.x = v.x * inv; o.y = v.y * inv;
  *(v2f*)(buf + off) = o;                // next hop's input
  v2f r = *(v2f*)(res + off);
  r.x += o.x; r.y += o.y;
  *(v2f*)(res + off) = r;
}

// ---------------------------------------------------------------------------
extern "C" void kernel_launch(void* const* d_in, const int* in_sizes, int n_in,
                              void* d_out, int out_size, void* d_ws, size_t ws_size,
                              hipStream_t stream) {
  (void)n_in; (void)out_size; (void)ws_size;
  const float* user_emb     = (const float*)d_in[0];
  const float* entity_emb   = (const float*)d_in[1];
  const float* interact_emb = (const float*)d_in[2];
  const float* relation_emb = (const float*)d_in[3];
  const int*   edge_index   = (const int*)d_in[4];
  const int*   edge_type    = (const int*)d_in[5];
  const int*   user_index   = (const int*)d_in[6];
  const int*   item_index   = (const int*)d_in[7];
  const int*   interact_ty  = (const int*)d_in[8];
  const int n_hops = 2;  // reference N_HOPS (scalar lives on-device; fixed by spec)

  const int nU  = in_sizes[0] / DEMB;
  const int nE  = in_sizes[1] / DEMB;
  const int E   = in_sizes[5];
  const int NNZ = in_sizes[6];
  const int* head = edge_index;       // edge_index[0]
  const int* tail = edge_index + E;   // edge_index[1]

  size_t pE = (size_t)nE * DEMB, pU = (size_t)nU * DEMB;
  float* ws    = (float*)d_ws;
  float* entB0 = ws; ws += pE;
  float* entB1 = ws; ws += pE;
  float* usrB0 = ws; ws += pU;
  float* usrB1 = ws; ws += pU;
  float* mE  = ws; ws += nE;
  float* sE  = ws; ws += nE;
  float* mU  = ws; ws += nU;
  float* sU  = ws; ws += nU;
  float* ewE = ws; ws += E;
  float* ewU = ws; ws += NNZ;

  float* entRes = (float*)d_out;
  float* usrRes = entRes + pE;
  hipMemcpyAsync(entRes, entity_emb, pE * sizeof(float), hipMemcpyDeviceToDevice, stream);
  hipMemcpyAsync(usrRes, user_emb,  pU * sizeof(float), hipMemcpyDeviceToDevice, stream);

  const int B = 256;
  auto cdiv = [](long long a, long long b) { return (int)((a + b - 1) / b); };

  const float* entIn = entity_emb;
  const float* usrIn = user_emb;
  float* entBufs[2] = {entB0, entB1};
  float* usrBufs[2] = {usrB0, usrB1};

  for (int hop = 0; hop < n_hops; ++hop) {
    float* entAgg = entBufs[hop & 1];
    float* usrAgg = usrBufs[hop & 1];
    hipMemsetAsync(entAgg, 0, pE * sizeof(float), stream);
    hipMemsetAsync(usrAgg, 0, pU * sizeof(float), stream);
    hipMemsetAsync(sE, 0, nE * sizeof(float), stream);
    hipMemsetAsync(sU, 0, nU * sizeof(float), stream);
    fill_k<<<cdiv(nE, B), B, 0, stream>>>(mE, -INFINITY, nE);
    fill_k<<<cdiv(nU, B), B, 0, stream>>>(mU, -INFINITY, nU);

    // --- KG entity aggregation ---
    kg_edge_logits<<<cdiv((long long)cdiv(E, 16) * 32, B), B, 0, stream>>>(
        entIn, relation_emb, tail, edge_type, ewE, E);
    seg_max_k<<<cdiv(E, B), B, 0, stream>>>(ewE, head, mE, E);
    seg_expsum_k<<<cdiv(E, B), B, 0, stream>>>(ewE, head, mE, sE, E);
    scatter_weighted_rows<<<cdiv((long long)E * DEMB, B), B, 0, stream>>>(
        entIn, tail, head, ewE, sE, entAgg, E);

    // --- user aggregation (uses pre-update entIn/usrIn) ---
    user_edge_logits<<<cdiv((long long)cdiv(NNZ, 16) * 32, B), B, 0, stream>>>(
        entIn, usrIn, interact_emb, user_index, item_index, interact_ty, ewU, NNZ);
    seg_max_k<<<cdiv(NNZ, B), B, 0, stream>>>(ewU, user_index, mU, NNZ);
    seg_expsum_k<<<cdiv(NNZ, B), B, 0, stream>>>(ewU, user_index, mU, sU, NNZ);
    scatter_weighted_rows<<<cdiv((long long)NNZ * DEMB, B), B, 0, stream>>>(
        entIn, item_index, user_index, ewU, sU, usrAgg, NNZ);

    // --- normalize in place + residual ---
    l2norm_residual<<<cdiv((long long)nE * 32, B), B, 0, stream>>>(entAgg, entRes, nE);
    l2norm_residual<<<cdiv((long long)nU * 32, B), B, 0, stream>>>(usrAgg, usrRes, nU);

    entIn = entAgg;
    usrIn = usrAgg;
  }
}